// RQBottleneck_45758581572096
// MI455X (gfx1250) — compile-verified
//
#include <hip/hip_runtime.h>

#define EMBED  256
#define NTOK   8192      // 8 * 1024 tokens
#define KCODES 8192
#define KPAD   8193      // codebook rows incl. padding row
#define RQDEPTH 4

typedef float v2f __attribute__((ext_vector_type(2)));
typedef float v8f __attribute__((ext_vector_type(8)));

// One wave (32 threads) owns a 16-row tile and runs all 4 depths.
// WMMA f32 16x16x4: A = 16 rows x 4 dims of residual, B = 4 dims x 16 codes,
// C accumulates the 16x16 dot-product tile over D=256 (64 WMMA steps,
// issued as 32 b128 loads each feeding two K=4 steps).
__global__ __launch_bounds__(32) void rq_main(const float* __restrict__ x,
                                              const float* __restrict__ cbs,
                                              float* __restrict__ out,
                                              float* __restrict__ partials) {
  const int lane = threadIdx.x & 31;
  const int row  = lane & 15;   // row within tile (A) / code column (B)
  const int half = lane >> 4;   // which K-half of the WMMA fragment this lane feeds
  const int m0   = blockIdx.x * 16;

  float* codes = out + (size_t)NTOK * EMBED + 1;   // after out tensor + loss scalar

  // Residual fragments (start as x). af[s] covers dims [8s+4*half, 8s+4*half+3].
  const float* xrow = x + (size_t)(m0 + row) * EMBED + 4 * half;
  float4 af[32];
#pragma unroll
  for (int s = 0; s < 32; ++s)
    af[s] = *(const float4*)(xrow + 8 * s);

  float lossacc = 0.f;

#pragma unroll 1
  for (int d = 0; d < RQDEPTH; ++d) {
    const float* cb = cbs + (size_t)d * KPAD * EMBED;

    float best[8];
    int   bidx[8];
#pragma unroll
    for (int j = 0; j < 8; ++j) { best[j] = 3.4e38f; bidx[j] = 0; }

#pragma unroll 1
    for (int t = 0; t < KCODES / 16; ++t) {
      // Lane loads code row (16t + row); halves split the dim range like A.
      const float* cbn = cb + (size_t)(t * 16 + row) * EMBED + 4 * half;
      v8f  c  = {0.f, 0.f, 0.f, 0.f, 0.f, 0.f, 0.f, 0.f};
      float cn = 0.f;                      // inline ||c||^2 partial for code (16t+row)
#pragma unroll
      for (int s = 0; s < 32; ++s) {
        float4 b = *(const float4*)(cbn + 8 * s);
        cn = fmaf(b.x, b.x, fmaf(b.y, b.y, fmaf(b.z, b.z, fmaf(b.w, b.w, cn))));
        v2f a0; a0[0] = af[s].x; a0[1] = af[s].y;
        v2f b0; b0[0] = b.x;     b0[1] = b.y;
        c = __builtin_amdgcn_wmma_f32_16x16x4_f32(false, a0, false, b0,
                                                  (short)0, c, false, false);
        v2f a1; a1[0] = af[s].z; a1[1] = af[s].w;
        v2f b1; b1[0] = b.z;     b1[1] = b.w;
        c = __builtin_amdgcn_wmma_f32_16x16x4_f32(false, a1, false, b1,
                                                  (short)0, c, false, false);
      }
      cn += __shfl_xor(cn, 16, 32);        // combine the two dim-halves of ||c||^2

      // score(m, n) = ||c_n||^2 - 2 * (r_m . c_n); ||r||^2 is argmin-invariant.
      // c[j] is (row j + 8*half, code 16t+row). Per-lane running min over its column.
      const int n = t * 16 + row;
#pragma unroll
      for (int j = 0; j < 8; ++j) {
        float val   = fmaf(-2.f, c[j], cn);
        bool better = (val < best[j]) || ((val == best[j]) && (n < bidx[j]));
        best[j] = better ? val : best[j];
        bidx[j] = better ? n   : bidx[j];
      }
    }

    // Cross-lane argmin over the 16 lanes of each half (columns 0..15 mod 16).
#pragma unroll
    for (int j = 0; j < 8; ++j) {
#pragma unroll
      for (int off = 8; off >= 1; off >>= 1) {
        float ov = __shfl_xor(best[j], off, 32);
        int   oi = __shfl_xor(bidx[j], off, 32);
        bool better = (ov < best[j]) || ((ov == best[j]) && (oi < bidx[j]));
        best[j] = better ? ov : best[j];
        bidx[j] = better ? oi : bidx[j];
      }
    }

    // Broadcast so every lane knows the winning index of *its* row (lane & 15).
    // Row j lives in lanes 0..15 (VGPR j), row j+8 in lanes 16..31.
    int myidx = 0;
#pragma unroll
    for (int j = 0; j < 8; ++j) {
      int i0 = __shfl(bidx[j], 0,  32);
      int i1 = __shfl(bidx[j], 16, 32);
      myidx = (row == j)     ? i0 : myidx;
      myidx = (row == j + 8) ? i1 : myidx;
    }

    if (lane < 16)
      codes[(size_t)(m0 + lane) * RQDEPTH + d] = (float)myidx;

    // residual -= cb[idx]; loss_d = mean(residual^2) since agg = x - residual.
    const float* q = cb + (size_t)myidx * EMBED + 4 * half;
#pragma unroll
    for (int s = 0; s < 32; ++s) {
      float4 b = *(const float4*)(q + 8 * s);
      af[s].x -= b.x; af[s].y -= b.y; af[s].z -= b.z; af[s].w -= b.w;
      lossacc = fmaf(af[s].x, af[s].x, fmaf(af[s].y, af[s].y,
                fmaf(af[s].z, af[s].z, fmaf(af[s].w, af[s].w, lossacc))));
    }
  }

  // Straight-through output: out = agg = x - residual_final.
  float* orow = out + (size_t)(m0 + row) * EMBED + 4 * half;
#pragma unroll
  for (int s = 0; s < 32; ++s) {
    float4 xv = *(const float4*)(xrow + 8 * s);
    float4 o;
    o.x = xv.x - af[s].x; o.y = xv.y - af[s].y;
    o.z = xv.z - af[s].z; o.w = xv.w - af[s].w;
    *(float4*)(orow + 8 * s) = o;
  }

  // Deterministic per-block loss partial (no float atomics).
#pragma unroll
  for (int off = 16; off >= 1; off >>= 1)
    lossacc += __shfl_xor(lossacc, off, 32);
  if (lane == 0) partials[blockIdx.x] = lossacc;
}

__global__ __launch_bounds__(32) void rq_loss_reduce(const float* __restrict__ partials,
                                                     float* __restrict__ out,
                                                     int nblocks) {
  int lane = threadIdx.x;
  float s = 0.f;
  for (int i = lane; i < nblocks; i += 32) s += partials[i];
#pragma unroll
  for (int off = 16; off >= 1; off >>= 1)
    s += __shfl_xor(s, off, 32);
  if (lane == 0)
    out[(size_t)NTOK * EMBED] =
        s * (0.25f / ((float)RQDEPTH * (float)NTOK * (float)EMBED));
}

extern "C" void kernel_launch(void* const* d_in, const int* in_sizes, int n_in,
                              void* d_out, int out_size, void* d_ws, size_t ws_size,
                              hipStream_t stream) {
  const float* x   = (const float*)d_in[0];   // [8,1024,256] f32
  const float* cbs = (const float*)d_in[1];   // [4,8193,256] f32
  float* out       = (float*)d_out;           // out | loss | codes (as float)
  float* partials  = (float*)d_ws;            // 512 floats of scratch

  const int nblocks = NTOK / 16;              // 512 waves
  rq_main<<<nblocks, 32, 0, stream>>>(x, cbs, out, partials);
  rq_loss_reduce<<<1, 32, 0, stream>>>(partials, out, nblocks);
}